// LightweightConv1dBTC_53171695125227
// MI455X (gfx1250) — compile-verified
//
#include <hip/hip_runtime.h>

// ---------------------------------------------------------------------------
// LightweightConv1d (BTC), depthwise causal conv K=31 with softmax'd weights.
// out[b,t,c] = sum_k softmax(w)[h(c)][k] * x[b, t-30+k, c]
// Band-matrix GEMM per (16t x 16c) tile on the CDNA5 matrix pipe:
//   Out(16x16) = A_band(16x48) x Xwin(48x16)  via 12x V_WMMA_F32_16X16X4_F32
// x-window staged into LDS with async global->LDS loads (ASYNCcnt path).
// ---------------------------------------------------------------------------

typedef __attribute__((ext_vector_type(2))) float v2f;
typedef __attribute__((ext_vector_type(8))) float v8f;
typedef __attribute__((ext_vector_type(4))) int   v4i;
typedef __attribute__((address_space(3))) v4i     lds_v4i;

#define Bn     16
#define Tn     2048
#define Cn     1024
#define Hn     16
#define Kn     31
#define PAD    30            // causal left padding
#define NKK    12            // 48/4 K-steps per 16x16 tile
#define TPW    2             // t-tiles per wave
#define WAVES  8
#define TSTRIP (WAVES * TPW * 16)     // 256 t-rows per block
#define NROWS  (TSTRIP + 32)          // 288 staged rows (window overhang)
#define NCHNK  (NROWS * 4)            // 16B chunks to stage per block

#if defined(__has_builtin)
# if __has_builtin(__builtin_amdgcn_global_load_async_to_lds_b128)
#  define HAVE_ASYNC_LDS 1
# endif
#endif
#ifndef HAVE_ASYNC_LDS
# define HAVE_ASYNC_LDS 0
#endif

// --- tiny pre-kernel: softmax over K for each of the 16 heads ---------------
__global__ void lwconv_softmax_w(const float* __restrict__ w,
                                 float* __restrict__ wsm) {
    int h = threadIdx.x;
    if (h < Hn) {
        float m = -1e30f;
        for (int k = 0; k < Kn; ++k) m = fmaxf(m, w[h * Kn + k]);
        float s = 0.0f;
        for (int k = 0; k < Kn; ++k) s += __expf(w[h * Kn + k] - m);
        float inv = 1.0f / s;
        for (int k = 0; k < Kn; ++k)
            wsm[h * Kn + k] = __expf(w[h * Kn + k] - m) * inv;
    }
}

// --- main kernel ------------------------------------------------------------
// Block = 256 threads = 8 waves; covers a 256-t strip of one 16-channel tile.
// grid.x = B * (T/256), grid.y = C/16.
__global__ __launch_bounds__(256) void lwconv_wmma(
    const float* __restrict__ x,
    const float* __restrict__ wsm,
    float* __restrict__ out) {

    __shared__ float xs[NROWS * 16];   // staged x-window, row-major [row][ch]
    __shared__ float wsh[32];          // head weights, padded (wsh[31] = 0)

    const int tid  = threadIdx.x;
    const int lane = tid & 31;
    const int wave = tid >> 5;
    const int half = lane >> 4;        // 0: lanes 0-15, 1: lanes 16-31
    const int l16  = lane & 15;

    const int b      = blockIdx.x >> 3;            // 8 strips per batch
    const int tstrip = (blockIdx.x & 7) * TSTRIP;
    const int c0     = blockIdx.y * 16;
    const int h      = c0 >> 6;                    // 64 channels per head

    const long base = (long)b * Tn * (long)Cn;

    // -- Stage head weights (padded so d&31 is always a valid LDS index). ----
    if (tid < 32)
        wsh[tid] = (tid < Kn) ? wsm[h * Kn + tid] : 0.0f;

    // -- Stage x-window rows [tstrip-30, tstrip+257] into LDS. ---------------
    // Each 16B chunk: row = ch>>2, part = ch&3. OOB t-rows are zero-filled.
    for (int ch = tid; ch < NCHNK; ch += 256) {
        const int row  = ch >> 2;
        const int part = ch & 3;
        const int trow = tstrip - PAD + row;
        float* ldst = &xs[row * 16 + part * 4];
        if (trow >= 0 && trow < Tn) {
            const float* gsrc = x + base + (long)trow * Cn + c0 + part * 4;
#if HAVE_ASYNC_LDS
            __builtin_amdgcn_global_load_async_to_lds_b128(
                (v4i*)gsrc, (lds_v4i*)ldst, 0, 0);
#else
            *(float4*)ldst = *(const float4*)gsrc;
#endif
        } else {
            *(float4*)ldst = make_float4(0.f, 0.f, 0.f, 0.f);
        }
    }

#if HAVE_ASYNC_LDS
# if defined(__has_builtin) && __has_builtin(__builtin_amdgcn_s_wait_asynccnt)
    __builtin_amdgcn_s_wait_asynccnt(0);
# else
    asm volatile("s_wait_asynccnt 0x0" ::: "memory");
# endif
#endif
    __syncthreads();

    // -- Build A-band fragments once per wave (branch-free: ds_load+select). -
    // 32-bit A 16x4 layout: lane(M)=l16; VGPR0 = K0|K2, VGPR1 = K1|K3.
    // A[i][j] = w[h][j-i] for 0<=j-i<=30, else 0.
    v2f afrag[NKK];
#pragma unroll
    for (int kk = 0; kk < NKK; ++kk) {
        int j0 = kk * 4 + half * 2;
        int d0 = j0 - l16;
        int d1 = d0 + 1;
        float v0 = wsh[d0 & 31];
        float v1 = wsh[d1 & 31];
        v2f a;
        a[0] = ((unsigned)d0 < (unsigned)Kn) ? v0 : 0.0f;
        a[1] = ((unsigned)d1 < (unsigned)Kn) ? v1 : 0.0f;
        afrag[kk] = a;
    }

    // -- Two 16x16 tiles per wave, 12 chained WMMAs each. --------------------
    // B 4x16 layout (K x N): VGPR0 = row K0|K2, VGPR1 = row K1|K3; N = l16.
#pragma unroll
    for (int tt = 0; tt < TPW; ++tt) {
        const int tile  = wave * TPW + tt;
        const int t0    = tstrip + tile * 16;
        const int lbase = tile * 16;           // xs row of window j=0

        v8f acc = {};
#pragma unroll
        for (int kk = 0; kk < NKK; ++kk) {
            const int lrow = lbase + kk * 4 + half * 2;
            v2f bfrag;
            bfrag[0] = xs[lrow * 16 + l16];
            bfrag[1] = xs[(lrow + 1) * 16 + l16];
            acc = __builtin_amdgcn_wmma_f32_16x16x4_f32(
                      false, afrag[kk], false, bfrag,
                      (short)0, acc, false, false);
        }

        // Store D: VGPR r holds row (r + half*8), col = l16.
#pragma unroll
        for (int r = 0; r < 8; ++r) {
            const int m = r + half * 8;
            out[base + (long)(t0 + m) * Cn + c0 + l16] = acc[r];
        }
    }
}

extern "C" void kernel_launch(void* const* d_in, const int* in_sizes, int n_in,
                              void* d_out, int out_size, void* d_ws, size_t ws_size,
                              hipStream_t stream) {
    const float* x = (const float*)d_in[0];   // (B, T, C) fp32
    const float* w = (const float*)d_in[1];   // (H, 1, K) fp32
    float* out = (float*)d_out;               // (B, T, C) fp32
    float* wsm = (float*)d_ws;                // H*K softmax'd weights

    hipLaunchKernelGGL(lwconv_softmax_w, dim3(1), dim3(32), 0, stream, w, wsm);

    dim3 grid(Bn * (Tn / TSTRIP), Cn / 16);   // 128 x 64 blocks
    hipLaunchKernelGGL(lwconv_wmma, grid, dim3(256), 0, stream, x, wsm, out);
}